// Discriminator_61598420959603
// MI455X (gfx1250) — compile-verified
//
#include <hip/hip_runtime.h>

typedef __attribute__((ext_vector_type(2))) float v2f;
typedef __attribute__((ext_vector_type(8))) float v8f;

#define L_SND 882000
#define FRAME 44100
#define HALF  22050
#define TSTEPS 2048

#define NC      (FRAME / 4)          // 11025 real K-chunks of 4
#define PER     1379                 // chunks per wave (8 * 1379 = 11032 >= NC)

// ---------------------------------------------------------------------------
// Kernel 1: gi[t][n] = sum_j |padded[alpha[t]+j]| * gru_w_ih[n][j]  (+ b_ih)
// via V_WMMA_F32_16X16X4_F32. 128 blocks x 16 rows; 8 waves split K=44100.
// Branchless edge handling (clamp + cndmask). Every wave runs a constant
// PER-iteration loop (tail chunks masked by zeroing B), so unroll-4 applies
// and ~12 loads stay in flight between WMMA groups.
// ---------------------------------------------------------------------------
__global__ __launch_bounds__(256) void k_gru_gemm(
    const float* __restrict__ sound, const int* __restrict__ alpha,
    const float* __restrict__ gw, const float* __restrict__ gb_ih,
    float* __restrict__ gi)
{
    const int tile = blockIdx.x;          // 0..127, 16 timesteps each
    const int tid  = threadIdx.x;
    const int wave = tid >> 5;
    const int lane = tid & 31;
    const int m    = lane & 15;           // row within tile (A) / column (B)
    const int koff = (lane >> 4) << 1;    // 0 or 2 within K-chunk of 4

    const int row   = tile * 16 + m;
    const int abase = alpha[row] - HALF;  // padded index -> sound index (int32)

    const bool nvalid = (m < 9);          // only 9 GRU weight rows
    const float* wrow = gw + (nvalid ? m * FRAME : 0);

    const int c0 = wave * PER;

    v8f acc = {0.f, 0.f, 0.f, 0.f, 0.f, 0.f, 0.f, 0.f};

#pragma unroll 4
    for (int i = 0; i < PER; ++i) {
        const int  c      = c0 + i;
        const bool kvalid = (c < NC);          // tail chunks contribute zero
        const int  k      = c * 4 + koff;
        const int  wk     = kvalid ? k : 0;    // clamped weight index

        const int p0 = abase + k;
        const int p1 = p0 + 1;
        const bool in0 = (unsigned)p0 < (unsigned)L_SND;
        const bool in1 = (unsigned)p1 < (unsigned)L_SND;
        const int  q0  = in0 ? p0 : 0;         // clamped, always in-bounds
        const int  q1  = in1 ? p1 : 0;

        const float s0 = sound[q0];            // unconditional loads
        const float s1 = sound[q1];
        const v2f  wv  = *(const v2f*)(wrow + wk);   // 8B-aligned pair

        const bool bval = nvalid && kvalid;
        v2f a, b;
        a.x = in0 ? __builtin_fabsf(s0) : 0.f;
        a.y = in1 ? __builtin_fabsf(s1) : 0.f;
        b.x = bval ? wv.x : 0.f;
        b.y = bval ? wv.y : 0.f;

        acc = __builtin_amdgcn_wmma_f32_16x16x4_f32(
                  false, a, false, b, (short)0, acc, false, false);
    }

    // cross-wave reduction of the 16x16 accumulators
    __shared__ float red[8 * 256];
#pragma unroll
    for (int v = 0; v < 8; ++v)
        red[wave * 256 + v * 32 + lane] = acc[v];
    __syncthreads();

    // C layout: VGPR v, lanes 0-15 -> (M=v, N=lane); lanes 16-31 -> (M=v+8)
    const int M = tid >> 4, N = tid & 15;
    float s = 0.f;
#pragma unroll
    for (int w = 0; w < 8; ++w)
        s += red[w * 256 + (M & 7) * 32 + (M >> 3) * 16 + N];
    if (N < 9)
        gi[(tile * 16 + M) * 9 + N] = s + gb_ih[N];
}

// ---------------------------------------------------------------------------
// Kernel 2: per-timestep GRU gates -> conv2 -> relu -> conv3 -> linear(17->10)
// ---------------------------------------------------------------------------
__global__ __launch_bounds__(256) void k_features(
    const int* __restrict__ alpha,
    const float* __restrict__ gi, const float* __restrict__ gb_hh,
    const float* __restrict__ w2, const float* __restrict__ b2,
    const float* __restrict__ w3, const float* __restrict__ b3,
    const float* __restrict__ lw, const float* __restrict__ lb,
    float* __restrict__ xin)
{
    const int t = blockIdx.x * blockDim.x + threadIdx.x;
    if (t >= TSTEPS) return;

    float g[9];
#pragma unroll
    for (int i = 0; i < 9; ++i) g[i] = gi[t * 9 + i];

    // GRU (h_prev = 0): h = (1-z) * tanh(gn + r*bhh_n)
    float h[3];
#pragma unroll
    for (int i = 0; i < 3; ++i) {
        float r  = 1.f / (1.f + expf(-(g[i]     + gb_hh[i])));
        float z  = 1.f / (1.f + expf(-(g[i + 3] + gb_hh[i + 3])));
        float nn = tanhf(g[i + 6] + r * gb_hh[i + 6]);
        h[i] = (1.f - z) * nn;
    }

    // conv2: (1,3) -> (32,3), kernel 3, pad 1, relu
    float y[32][3];
#pragma unroll
    for (int c = 0; c < 32; ++c) {
#pragma unroll
        for (int p = 0; p < 3; ++p) {
            float acc = b2[c];
#pragma unroll
            for (int k = 0; k < 3; ++k) {
                int q = p + k - 1;
                float xv = (q >= 0 && q < 3) ? h[q] : 0.f;
                acc += w2[c * 3 + k] * xv;
            }
            y[c][p] = acc > 0.f ? acc : 0.f;
        }
    }

    // conv3: (32,3) -> (16,1), kernel 3, pad 0
    float feat[16];
#pragma unroll
    for (int o = 0; o < 16; ++o) {
        float acc = b3[o];
        for (int c = 0; c < 32; ++c) {
#pragma unroll
            for (int k = 0; k < 3; ++k)
                acc += w3[(o * 32 + c) * 3 + k] * y[c][k];
        }
        feat[o] = acc;
    }

    const float iv = (t == 0) ? (float)alpha[0]
                              : (float)(alpha[t] - alpha[t - 1]);

    // linear 17 -> 10  (input = [interval, feat0..feat15])
#pragma unroll
    for (int o = 0; o < 10; ++o) {
        float acc = lb[o] + lw[o * 17] * iv;
#pragma unroll
        for (int i = 0; i < 16; ++i)
            acc += lw[o * 17 + 1 + i] * feat[i];
        xin[t * 10 + o] = acc;
    }
}

// ---------------------------------------------------------------------------
// Kernel 3: sequential LSTM (H=10, 2048 steps) + head. One 64-thread block:
// lanes 0-39 each own one gate row (weights in registers), h/c in LDS.
// ---------------------------------------------------------------------------
__global__ __launch_bounds__(64) void k_lstm_head(
    const float* __restrict__ xin,
    const float* __restrict__ wih, const float* __restrict__ whh,
    const float* __restrict__ bih, const float* __restrict__ bhh,
    const float* __restrict__ l1w, const float* __restrict__ l1b,
    const float* __restrict__ l2w, const float* __restrict__ l2b,
    float* __restrict__ out)
{
    __shared__ float h[10], cc[10], gbuf[40], ybuf[32];
    const int j = threadIdx.x;

    float rih[10], rhh[10], bj = 0.f;
    if (j < 40) {
#pragma unroll
        for (int i = 0; i < 10; ++i) {
            rih[i] = wih[j * 10 + i];
            rhh[i] = whh[j * 10 + i];
        }
        bj = bih[j] + bhh[j];
    }
    if (j < 10) { h[j] = 0.f; cc[j] = 0.f; }
    __syncthreads();

    for (int t = 0; t < TSTEPS; ++t) {
        if (j < 40) {
            float g = bj;
#pragma unroll
            for (int i = 0; i < 10; ++i) g += xin[t * 10 + i] * rih[i];
#pragma unroll
            for (int i = 0; i < 10; ++i) g += h[i] * rhh[i];
            gbuf[j] = g;
        }
        __syncthreads();
        if (j < 10) {
            float ig = 1.f / (1.f + expf(-gbuf[j]));
            float fg = 1.f / (1.f + expf(-gbuf[j + 10]));
            float gg = tanhf(gbuf[j + 20]);
            float og = 1.f / (1.f + expf(-gbuf[j + 30]));
            float c2 = fg * cc[j] + ig * gg;
            cc[j] = c2;
            h[j]  = og * tanhf(c2);
        }
        __syncthreads();
    }

    if (j < 32) {
        float yv = l1b[j];
#pragma unroll
        for (int i = 0; i < 10; ++i) yv += h[i] * l1w[j * 10 + i];
        ybuf[j] = yv > 0.f ? yv : 0.f;
    }
    __syncthreads();
    if (j == 0) {
        float yv = l2b[0];
#pragma unroll
        for (int i = 0; i < 32; ++i) yv += ybuf[i] * l2w[i];
        out[0] = 1.f / (1.f + expf(-yv));
    }
}

// ---------------------------------------------------------------------------
extern "C" void kernel_launch(void* const* d_in, const int* in_sizes, int n_in,
                              void* d_out, int out_size, void* d_ws, size_t ws_size,
                              hipStream_t stream) {
    (void)in_sizes; (void)n_in; (void)out_size; (void)ws_size;

    const float* sound  = (const float*)d_in[0];
    const int*   alpha  = (const int*)  d_in[1];
    const float* gw_ih  = (const float*)d_in[2];
    const float* gb_ih  = (const float*)d_in[3];
    const float* gb_hh  = (const float*)d_in[4];
    const float* w2     = (const float*)d_in[5];
    const float* b2     = (const float*)d_in[6];
    const float* w3     = (const float*)d_in[7];
    const float* b3     = (const float*)d_in[8];
    const float* lw     = (const float*)d_in[9];
    const float* lb     = (const float*)d_in[10];
    const float* lwih   = (const float*)d_in[11];
    const float* lwhh   = (const float*)d_in[12];
    const float* lbih   = (const float*)d_in[13];
    const float* lbhh   = (const float*)d_in[14];
    const float* l1w    = (const float*)d_in[15];
    const float* l1b    = (const float*)d_in[16];
    const float* l2w    = (const float*)d_in[17];
    const float* l2b    = (const float*)d_in[18];

    float* gi   = (float*)d_ws;          // 2048*9 f32
    float* xin  = gi + TSTEPS * 9;       // 2048*10 f32

    k_gru_gemm<<<dim3(TSTEPS / 16), dim3(256), 0, stream>>>(
        sound, alpha, gw_ih, gb_ih, gi);

    k_features<<<dim3(TSTEPS / 256), dim3(256), 0, stream>>>(
        alpha, gi, gb_hh, w2, b2, w3, b3, lw, lb, xin);

    k_lstm_head<<<dim3(1), dim3(64), 0, stream>>>(
        xin, lwih, lwhh, lbih, lbhh, l1w, l1b, l2w, l2b, (float*)d_out);
}